// Transformer_13477607374929
// MI455X (gfx1250) — compile-verified
//
#include <hip/hip_runtime.h>
#include <cstdint>

// Model dims (fixed by the reference)
#define Dm   768
#define TD3  2304          // 3*D
#define Lm   12
#define Hm   4
#define HDm  192
#define Bm   2
#define Tm   1024
#define Vm   50257
#define MROWS (Bm*Tm)      // 2048

typedef __attribute__((ext_vector_type(2))) float v2f;
typedef __attribute__((ext_vector_type(8))) float v8f;

#define GF_RELU   1
#define GF_RESID  2
#define GF_NBOUND 4

// ---------------------------------------------------------------- embedding
__global__ void embed_kernel(const int* __restrict__ tokens,
                             const float* __restrict__ tok_emb,
                             const float* __restrict__ pos_emb,
                             float* __restrict__ x) {
  int row = blockIdx.x;            // b*T + t
  int t   = row & (Tm - 1);
  int tok = tokens[row];
  const float* te = tok_emb + (size_t)tok * Dm;
  const float* pe = pos_emb + (size_t)t * Dm;
  float* xp = x + (size_t)row * Dm;
  for (int d = threadIdx.x; d < Dm; d += blockDim.x)
    xp[d] = te[d] + pe[d];
}

// ---------------------------------------------------------------- layernorm
__global__ void layernorm_kernel(const float* __restrict__ x,
                                 const float* __restrict__ scale,
                                 const float* __restrict__ shift,
                                 float* __restrict__ out) {
  int row = blockIdx.x;
  const float* xp = x + (size_t)row * Dm;
  float* op = out + (size_t)row * Dm;
  __shared__ float r1[256], r2[256];
  int tid = threadIdx.x;
  float s = 0.f, s2 = 0.f;
  for (int d = tid; d < Dm; d += 256) { float v = xp[d]; s += v; s2 += v * v; }
  r1[tid] = s; r2[tid] = s2; __syncthreads();
  for (int k = 128; k > 0; k >>= 1) {
    if (tid < k) { r1[tid] += r1[tid + k]; r2[tid] += r2[tid + k]; }
    __syncthreads();
  }
  float mu  = r1[0] * (1.0f / Dm);
  float var = r2[0] * (1.0f / Dm) - mu * mu;
  float inv = rsqrtf(var + 1e-5f);
  for (int d = tid; d < Dm; d += 256)
    op[d] = (xp[d] - mu) * inv * scale[d] + shift[d];
}

// ------------------------------------------------ generic GEMM: C = A*W^T (+bias, relu, resid)
// A[M,K] row-major, W[N,K] row-major (so C[m,n] = sum_k A[m,k]*W[n,k]).
// One wave -> one 16x16 C tile. 4 waves/block -> 64 output cols.
__global__ void gemm_wmma_kernel(const float* __restrict__ A,
                                 const float* __restrict__ W,
                                 const float* __restrict__ bias,
                                 const float* __restrict__ resid,
                                 float* __restrict__ C,
                                 int M, int N, int K, int flags) {
  int lane  = threadIdx.x & 31;
  int wave  = threadIdx.x >> 5;
  int mBase = blockIdx.y << 4;
  int nBase = (blockIdx.x << 6) + (wave << 4);
  int half  = lane >> 4;                // 0: K=0,1  1: K=2,3
  int lid   = lane & 15;
  int kOff  = half << 1;
  int m = mBase + lid;
  int n = nBase + lid;
  int nw = n;
  if (flags & GF_NBOUND) nw = (n < N) ? n : (N - 1);
  const float* ap = A + (size_t)m  * K + kOff;
  const float* wp = W + (size_t)nw * K + kOff;
  v8f acc = {};
  for (int k = 0; k < K; k += 4) {
    v2f a = *(const v2f*)(ap + k);
    v2f b = *(const v2f*)(wp + k);
    acc = __builtin_amdgcn_wmma_f32_16x16x4_f32(false, a, false, b,
                                                (short)0, acc, false, false);
  }
  bool nOK = (!(flags & GF_NBOUND)) || (n < N);
  float bn = nOK ? bias[n] : 0.f;
#pragma unroll
  for (int r = 0; r < 8; ++r) {
    int mm = mBase + r + (half << 3);
    float v = acc[r] + bn;
    if (flags & GF_RELU)  v = fmaxf(v, 0.f);
    if (flags & GF_RESID) v += resid[(size_t)mm * N + n];
    if (nOK) C[(size_t)mm * N + n] = v;
  }
}

// ------------------------------------------------ attention scores: S = QK^T/sqrt(hd), causal
__global__ void attn_scores_kernel(const float* __restrict__ qkv,
                                   float* __restrict__ S) {
  int lane = threadIdx.x & 31;
  int wave = threadIdx.x >> 5;
  int bh = blockIdx.z;               // b*H + h
  int b = bh >> 2, h = bh & 3;
  int qBase = blockIdx.y << 4;
  int kBase = (blockIdx.x << 6) + (wave << 4);
  int half = lane >> 4, lid = lane & 15, kOff = half << 1;
  int kc = kBase + lid;

  if (kBase > qBase + 15) {          // fully masked tile: write -inf-ish, skip compute
#pragma unroll
    for (int r = 0; r < 8; ++r) {
      int q = qBase + r + (half << 3);
      S[((size_t)bh * Tm + q) * Tm + kc] = -1e30f;
    }
    return;
  }
  const float* qp = qkv + (size_t)(b * Tm + qBase + lid) * TD3 + h * HDm + kOff;
  const float* kp = qkv + (size_t)(b * Tm + kBase + lid) * TD3 + Dm + h * HDm + kOff;
  v8f acc = {};
  for (int d = 0; d < HDm; d += 4) {
    v2f a  = *(const v2f*)(qp + d);
    v2f bb = *(const v2f*)(kp + d);
    acc = __builtin_amdgcn_wmma_f32_16x16x4_f32(false, a, false, bb,
                                                (short)0, acc, false, false);
  }
  const float scl = 0.07216878364870323f;   // 1/sqrt(192)
#pragma unroll
  for (int r = 0; r < 8; ++r) {
    int q = qBase + r + (half << 3);
    float v = (kc <= q) ? acc[r] * scl : -1e30f;
    S[((size_t)bh * Tm + q) * Tm + kc] = v;
  }
}

// ---------------------------------------------------------------- row softmax over T
__global__ void softmax_kernel(float* __restrict__ S) {
  int row = blockIdx.x;
  float* p = S + (size_t)row * Tm;
  __shared__ float red[256];
  int tid = threadIdx.x;
  float vmax = -1e30f;
  for (int i = tid; i < Tm; i += 256) vmax = fmaxf(vmax, p[i]);
  red[tid] = vmax; __syncthreads();
  for (int k = 128; k > 0; k >>= 1) {
    if (tid < k) red[tid] = fmaxf(red[tid], red[tid + k]);
    __syncthreads();
  }
  float m = red[0]; __syncthreads();
  float sum = 0.f;
  for (int i = tid; i < Tm; i += 256) { float e = __expf(p[i] - m); p[i] = e; sum += e; }
  red[tid] = sum; __syncthreads();
  for (int k = 128; k > 0; k >>= 1) {
    if (tid < k) red[tid] += red[tid + k];
    __syncthreads();
  }
  float inv = 1.f / red[0];
  for (int i = tid; i < Tm; i += 256) p[i] *= inv;
}

// ------------------------------------------------ y = P*V, accumulated into residual x
__global__ void attn_av_kernel(const float* __restrict__ S,
                               const float* __restrict__ qkv,
                               float* __restrict__ x) {
  int lane = threadIdx.x & 31;
  int wave = threadIdx.x >> 5;
  int bh = blockIdx.z; int b = bh >> 2, h = bh & 3;
  int qBase = blockIdx.y << 4;
  int dTile = (blockIdx.x << 2) + wave;   // 0..11
  int dBase = dTile << 4;
  int half = lane >> 4, lid = lane & 15, kOff = half << 1;
  const float* sp = S + ((size_t)bh * Tm + qBase + lid) * Tm + kOff;
  const float* vp = qkv + (size_t)(b * Tm) * TD3 + 2 * Dm + h * HDm + dBase + lid;
  v8f acc = {};
  int jEnd = qBase + 16;                  // causal: weights beyond are exactly 0
  for (int j = 0; j < jEnd; j += 4) {
    v2f a = *(const v2f*)(sp + j);
    v2f bb;
    bb.x = vp[(size_t)(j + kOff)     * TD3];
    bb.y = vp[(size_t)(j + kOff + 1) * TD3];
    acc = __builtin_amdgcn_wmma_f32_16x16x4_f32(false, a, false, bb,
                                                (short)0, acc, false, false);
  }
  int n = h * HDm + dBase + lid;
#pragma unroll
  for (int r = 0; r < 8; ++r) {
    int t = qBase + r + (half << 3);
    x[(size_t)(b * Tm + t) * Dm + n] += acc[r];
  }
}

// ---------------------------------------------------------------- driver
extern "C" void kernel_launch(void* const* d_in, const int* in_sizes, int n_in,
                              void* d_out, int out_size, void* d_ws, size_t ws_size,
                              hipStream_t stream) {
  (void)in_sizes; (void)n_in; (void)out_size; (void)ws_size;
  const int*   tokens  = (const int*)d_in[0];
  const float* tok_emb = (const float*)d_in[1];
  const float* pos_emb = (const float*)d_in[2];
  const float* ln1_s   = (const float*)d_in[3];
  const float* ln1_b   = (const float*)d_in[4];
  const float* qkv_w   = (const float*)d_in[5];
  const float* qkv_b   = (const float*)d_in[6];
  const float* ln2_s   = (const float*)d_in[7];
  const float* ln2_b   = (const float*)d_in[8];
  const float* fc_w    = (const float*)d_in[9];
  const float* fc_b    = (const float*)d_in[10];
  const float* proj_w  = (const float*)d_in[11];
  const float* proj_b  = (const float*)d_in[12];
  const float* lnf_s   = (const float*)d_in[13];
  const float* lnf_b   = (const float*)d_in[14];
  const float* lm_w    = (const float*)d_in[15];
  const float* lm_b    = (const float*)d_in[16];
  float* out = (float*)d_out;

  float* ws   = (float*)d_ws;
  float* x    = ws;                               // [2048, 768]
  float* hbuf = x    + (size_t)MROWS * Dm;        // [2048, 768]
  float* qkv  = hbuf + (size_t)MROWS * Dm;        // [2048, 2304]
  float* mbuf = qkv  + (size_t)MROWS * 3 * Dm;    // [2048, 3072]
  float* Sbuf = mbuf + (size_t)MROWS * 4 * Dm;    // [8, 1024, 1024]

  dim3 blk(128);
  embed_kernel<<<dim3(MROWS), dim3(256), 0, stream>>>(tokens, tok_emb, pos_emb, x);

  for (int l = 0; l < Lm; ++l) {
    layernorm_kernel<<<MROWS, 256, 0, stream>>>(x, ln1_s + l * Dm, ln1_b + l * Dm, hbuf);
    gemm_wmma_kernel<<<dim3((3 * Dm) / 64, MROWS / 16), blk, 0, stream>>>(
        hbuf, qkv_w + (size_t)l * 3 * Dm * Dm, qkv_b + (size_t)l * 3 * Dm,
        nullptr, qkv, MROWS, 3 * Dm, Dm, 0);
    attn_scores_kernel<<<dim3(Tm / 64, Tm / 16, Bm * Hm), blk, 0, stream>>>(qkv, Sbuf);
    softmax_kernel<<<Bm * Hm * Tm, 256, 0, stream>>>(Sbuf);
    attn_av_kernel<<<dim3(HDm / 64, Tm / 16, Bm * Hm), blk, 0, stream>>>(Sbuf, qkv, x);
    layernorm_kernel<<<MROWS, 256, 0, stream>>>(x, ln2_s + l * Dm, ln2_b + l * Dm, hbuf);
    gemm_wmma_kernel<<<dim3((4 * Dm) / 64, MROWS / 16), blk, 0, stream>>>(
        hbuf, fc_w + (size_t)l * 4 * Dm * Dm, fc_b + (size_t)l * 4 * Dm,
        nullptr, mbuf, MROWS, 4 * Dm, Dm, GF_RELU);
    gemm_wmma_kernel<<<dim3(Dm / 64, MROWS / 16), blk, 0, stream>>>(
        mbuf, proj_w + (size_t)l * Dm * 4 * Dm, proj_b + (size_t)l * Dm,
        x, x, MROWS, Dm, 4 * Dm, GF_RESID);
  }

  layernorm_kernel<<<MROWS, 256, 0, stream>>>(x, lnf_s, lnf_b, hbuf);
  gemm_wmma_kernel<<<dim3((Vm + 63) / 64, MROWS / 16), blk, 0, stream>>>(
      hbuf, lm_w, lm_b, nullptr, out, MROWS, Vm, Dm, GF_NBOUND);
}